// QuantumKernelFeedForward_65481071408021
// MI455X (gfx1250) — compile-verified
//
#include <hip/hip_runtime.h>
#include <stdint.h>

typedef __attribute__((ext_vector_type(2))) float v2f;
typedef __attribute__((ext_vector_type(8))) float v8f;

#define M_ROWS 4096   // BATCH*SEQ
#define EMBED  512
#define FFN    2048

// ---------------------------------------------------------------------------
// CDNA5 async copy: global -> LDS, 16 bytes per lane, tracked by ASYNCcnt.
// ---------------------------------------------------------------------------
__device__ __forceinline__ uint32_t lds_off(const void* p) {
    // generic pointers to LDS carry the LDS byte offset in their low 32 bits
    return (uint32_t)(uintptr_t)p;
}
__device__ __forceinline__ void async_b128(uint32_t dst_lds, const void* src_global) {
    asm volatile("global_load_async_to_lds_b128 %0, %1, off"
                 :: "v"(dst_lds), "v"((uint64_t)(uintptr_t)src_global)
                 : "memory");
}
__device__ __forceinline__ void wait_async0() {
#if __has_builtin(__builtin_amdgcn_s_wait_asynccnt)
    __builtin_amdgcn_s_wait_asynccnt(0);
#else
    asm volatile("s_wait_asynccnt 0" ::: "memory");
#endif
}

__device__ __forceinline__ v8f wmma4(v2f a, v2f b, v8f c) {
    return __builtin_amdgcn_wmma_f32_16x16x4_f32(false, a, false, b, (short)0, c,
                                                 false, false);
}

// ---------------------------------------------------------------------------
// Kernel 1: out = gram @ flat, gram[i][j] = |prod_w cos(0.5*(f_i[w]-f_j[w]))|
// Workgroup (256 thr = 8 waves) owns a 32-row block; each wave owns
// 2 M-tiles x 4 N-tiles (A-frag reused 4x, B-frag reused 2x). K chunked by 256
// with gram chunk built in LDS; f_j chunks double-buffered via async DMA.
// ---------------------------------------------------------------------------
#define K1_KC  256
#define K1_PAD 260   // stride mod 64 banks == 4 -> conflict-free column reads

__global__ __launch_bounds__(256)
void qkff_gram_matmul(const float* __restrict__ x, float* __restrict__ out) {
    __shared__ float Gs[32 * K1_PAD];    // gram chunk (33.3 KB)
    __shared__ float FJ[2][K1_KC * 4];   // double-buffered f_j (2 x 4 KB)
    __shared__ float FI[32 * 4];         // f_i for this row block

    const int tid  = threadIdx.x;
    const int lane = tid & 31;
    const int wave = tid >> 5;
    const int lr   = lane & 15;          // tile row / tile col within lane half
    const int hi   = lane >> 4;          // 0: K+0/K+1 ; 1: K+2/K+3 (A layout)
    const int mb   = blockIdx.x * 32;    // output row block
    const int nb   = wave * 64;          // 4 N-tiles per wave

    // prologue: async-stage f_i (32 rows) and f_j chunk 0 (rows 0..255)
    if (tid < 32) async_b128(lds_off(&FI[tid * 4]), &x[(size_t)(mb + tid) * EMBED]);
    async_b128(lds_off(&FJ[0][tid * 4]), &x[(size_t)tid * EMBED]);

    v8f a00 = {}, a01 = {}, a02 = {}, a03 = {};
    v8f a10 = {}, a11 = {}, a12 = {}, a13 = {};

    const int nchunks = M_ROWS / K1_KC;
    for (int c = 0; c < nchunks; ++c) {
        const int k0 = c * K1_KC;
        wait_async0();          // own async copies landed in LDS
        __syncthreads();        // everyone's landed; prev-chunk Gs reads done

        {   // build gram chunk: thread owns column jj=tid for all 32 rows
            const float* fj = &FJ[c & 1][tid * 4];
            const float fj0 = fj[0], fj1 = fj[1], fj2 = fj[2], fj3 = fj[3];
            #pragma unroll
            for (int i = 0; i < 32; ++i) {
                float c0 = __cosf(0.5f * (FI[i * 4 + 0] - fj0));
                float c1 = __cosf(0.5f * (FI[i * 4 + 1] - fj1));
                float c2 = __cosf(0.5f * (FI[i * 4 + 2] - fj2));
                float c3 = __cosf(0.5f * (FI[i * 4 + 3] - fj3));
                Gs[i * K1_PAD + tid] = fabsf(c0 * c1 * c2 * c3);
            }
        }
        // prefetch next f_j chunk into the other buffer (overlaps WMMA phase)
        if (c + 1 < nchunks)
            async_b128(lds_off(&FJ[(c + 1) & 1][tid * 4]),
                       &x[(size_t)(k0 + K1_KC + tid) * EMBED]);
        __syncthreads();

        // WMMA: A = gram slice (LDS), B = flat slice (L2-resident)
        const float* g0 = &Gs[lr * K1_PAD + hi * 2];
        const float* g1 = &Gs[(16 + lr) * K1_PAD + hi * 2];
        #pragma unroll 2
        for (int kk = 0; kk < K1_KC; kk += 4) {
            v2f fa0 = *(const v2f*)&g0[kk];              // ds_load_b64: K,K+1
            v2f fa1 = *(const v2f*)&g1[kk];
            const float* bp = &x[(size_t)(k0 + kk + hi * 2) * EMBED + nb + lr];
            v2f b0 = { bp[ 0], bp[ 0 + EMBED] };
            v2f b1 = { bp[16], bp[16 + EMBED] };
            v2f b2 = { bp[32], bp[32 + EMBED] };
            v2f b3 = { bp[48], bp[48 + EMBED] };
            a00 = wmma4(fa0, b0, a00);  a01 = wmma4(fa0, b1, a01);
            a02 = wmma4(fa0, b2, a02);  a03 = wmma4(fa0, b3, a03);
            a10 = wmma4(fa1, b0, a10);  a11 = wmma4(fa1, b1, a11);
            a12 = wmma4(fa1, b2, a12);  a13 = wmma4(fa1, b3, a13);
        }
    }

    // C/D layout: VGPR r <-> row r (lanes 0-15) / r+8 (lanes 16-31), col lane%16
    #pragma unroll
    for (int r = 0; r < 8; ++r) {
        size_t r0 = (size_t)(mb + r + hi * 8) * EMBED + nb + lr;
        size_t r1 = r0 + (size_t)16 * EMBED;
        out[r0 +  0] = a00[r];  out[r0 + 16] = a01[r];
        out[r0 + 32] = a02[r];  out[r0 + 48] = a03[r];
        out[r1 +  0] = a10[r];  out[r1 + 16] = a11[r];
        out[r1 + 32] = a12[r];  out[r1 + 48] = a13[r];
    }
}

// ---------------------------------------------------------------------------
// Kernels 2/3: C = [relu](A @ B + bias).  A: MxK, B: KxN, row-major.
// Workgroup = 32 rows x 256 cols; wave owns 2 M-tiles x 2 N-tiles.
// A row-block double-buffered in LDS via async DMA, K chunked by 128.
// ---------------------------------------------------------------------------
#define G_KC  128
#define G_PAD 132    // stride mod 64 == 4; 132*4 % 16 == 0 (b128 alignment)

template <bool RELU>
__global__ __launch_bounds__(256)
void qkff_gemm_bias(const float* __restrict__ A, const float* __restrict__ B,
                    const float* __restrict__ bias, float* __restrict__ C,
                    int M, int N, int K) {
    __shared__ float As[2][32 * G_PAD];   // 2 x 16.9 KB

    const int tid  = threadIdx.x;
    const int lane = tid & 31;
    const int wave = tid >> 5;
    const int lr   = lane & 15;
    const int hi   = lane >> 4;
    const int ngroups = N / 256;
    const int mb = (blockIdx.x / ngroups) * 32;
    const int n  = (blockIdx.x % ngroups) * 256 + wave * 32;

    // prologue: async-stage A[mb:mb+32][0:G_KC] into buffer 0
    #pragma unroll
    for (int idx = tid; idx < 32 * (G_KC / 4); idx += 256) {
        const int row = idx >> 5;          // idx / (G_KC/4)
        const int c4  = idx & 31;          // idx % (G_KC/4)
        async_b128(lds_off(&As[0][row * G_PAD + c4 * 4]),
                   &A[(size_t)(mb + row) * K + c4 * 4]);
    }

    v8f a00 = {}, a01 = {}, a10 = {}, a11 = {};

    const int nchunks = K / G_KC;
    for (int c = 0; c < nchunks; ++c) {
        const int k0 = c * G_KC;
        wait_async0();
        __syncthreads();
        // prefetch next A chunk into the other buffer (overlaps WMMA phase)
        if (c + 1 < nchunks) {
            #pragma unroll
            for (int idx = tid; idx < 32 * (G_KC / 4); idx += 256) {
                const int row = idx >> 5;
                const int c4  = idx & 31;
                async_b128(lds_off(&As[(c + 1) & 1][row * G_PAD + c4 * 4]),
                           &A[(size_t)(mb + row) * K + k0 + G_KC + c4 * 4]);
            }
        }
        const float* a0p = &As[c & 1][lr * G_PAD + hi * 2];
        const float* a1p = &As[c & 1][(16 + lr) * G_PAD + hi * 2];
        #pragma unroll 4
        for (int kk = 0; kk < G_KC; kk += 4) {
            v2f fa0 = *(const v2f*)&a0p[kk];
            v2f fa1 = *(const v2f*)&a1p[kk];
            const float* bp = &B[(size_t)(k0 + kk + hi * 2) * N + n + lr];
            v2f b0 = { bp[ 0], bp[ 0 + N] };
            v2f b1 = { bp[16], bp[16 + N] };
            a00 = wmma4(fa0, b0, a00);  a01 = wmma4(fa0, b1, a01);
            a10 = wmma4(fa1, b0, a10);  a11 = wmma4(fa1, b1, a11);
        }
    }

    const float bv0 = bias[n + lr];
    const float bv1 = bias[n + 16 + lr];
    #pragma unroll
    for (int r = 0; r < 8; ++r) {
        float v00 = a00[r] + bv0, v01 = a01[r] + bv1;
        float v10 = a10[r] + bv0, v11 = a11[r] + bv1;
        if (RELU) {
            v00 = fmaxf(v00, 0.0f);  v01 = fmaxf(v01, 0.0f);
            v10 = fmaxf(v10, 0.0f);  v11 = fmaxf(v11, 0.0f);
        }
        size_t r0 = (size_t)(mb + r + hi * 8) * N + n + lr;
        size_t r1 = r0 + (size_t)16 * N;
        C[r0] = v00;  C[r0 + 16] = v01;
        C[r1] = v10;  C[r1 + 16] = v11;
    }
}

// ---------------------------------------------------------------------------
extern "C" void kernel_launch(void* const* d_in, const int* in_sizes, int n_in,
                              void* d_out, int out_size, void* d_ws, size_t ws_size,
                              hipStream_t stream) {
    const float* x  = (const float*)d_in[0];   // (8,512,512)
    const float* W1 = (const float*)d_in[1];   // (512,2048)
    const float* b1 = (const float*)d_in[2];   // (2048,)
    const float* W2 = (const float*)d_in[3];   // (2048,512)
    const float* b2 = (const float*)d_in[4];   // (512,)
    float* y = (float*)d_out;                  // (8,512,512)

    float* out_buf = (float*)d_ws;                        // 4096 x 512
    float* h_buf   = out_buf + (size_t)M_ROWS * EMBED;    // 4096 x 2048

    qkff_gram_matmul<<<M_ROWS / 32, 256, 0, stream>>>(x, out_buf);

    qkff_gemm_bias<true><<<(M_ROWS / 32) * (FFN / 256), 256, 0, stream>>>(
        out_buf, W1, b1, h_buf, M_ROWS, FFN, EMBED);

    qkff_gemm_bias<false><<<(M_ROWS / 32) * (EMBED / 256), 256, 0, stream>>>(
        h_buf, W2, b2, y, M_ROWS, EMBED, FFN);
}